// SelectRegions_50560355009104
// MI455X (gfx1250) — compile-verified
//
#include <hip/hip_runtime.h>
#include <cstdint>

// ---------------- problem geometry ----------------
#define KCLS 20
#define HIN 480
#define WIN 640
#define NPIX (HIN * WIN)          // 307200
#define NBATCH 4
#define CCH 512
#define HF 30
#define WF 40
#define SLAB (CCH * HF * WF)      // 614400 floats per (n or node) slab
#define NB 5
#define COUNT_THRESH 10000
#define CHUNK 8192                // floats per TDM tile (32 KB LDS)
#define OUT_X_OFF 2050            // pool_x(2048) + NB + N
#define RESIZE_ELEMS ((size_t)20 * SLAB)   // 12,288,000
#define COPY_FLOATS ((size_t)4 * SLAB)     // 2,457,600 (feat verbatim)

// ws layout (ints): [0:20) cnt, [20:40) xmin, [40:60) xmax,
//                   [60:80) ymin, [80:100) ymax, [100:120) node params {x0,y0,iw,ih}x5
typedef unsigned int v4u __attribute__((ext_vector_type(4)));
typedef int          v4i __attribute__((ext_vector_type(4)));
typedef int          v8i __attribute__((ext_vector_type(8)));

// ---------------- stats init ----------------
__global__ void k_init(int* __restrict__ ws) {
  int t = threadIdx.x;
  if (t < KCLS) {
    ws[t]        = 0;
    ws[20 + t]   = 0x7fffffff;      // xmin
    ws[40 + t]   = (int)0x80000000; // xmax
    ws[60 + t]   = 0x7fffffff;      // ymin
    ws[80 + t]   = (int)0x80000000; // ymax
  }
}

// ---------------- fused argmax (batch N-1 only) + per-class bbox/count ----------------
__global__ void k_stats(const float* __restrict__ seg, int* __restrict__ ws) {
  __shared__ int s_cnt[KCLS], s_xmin[KCLS], s_xmax[KCLS], s_ymin[KCLS], s_ymax[KCLS];
  int t = threadIdx.x;
  if (t < KCLS) {
    s_cnt[t] = 0;
    s_xmin[t] = 0x7fffffff; s_xmax[t] = (int)0x80000000;
    s_ymin[t] = 0x7fffffff; s_ymax[t] = (int)0x80000000;
  }
  __syncthreads();
  int p = blockIdx.x * blockDim.x + t;
  if (p < NPIX) {
    const float* base = seg + (size_t)(NBATCH - 1) * KCLS * NPIX + p;
    float best = base[0];
    int bk = 0;
#pragma unroll
    for (int k = 1; k < KCLS; ++k) {
      float v = base[(size_t)k * NPIX];
      if (v > best) { best = v; bk = k; }   // strict > keeps first index (jnp.argmax)
    }
    int x = p % WIN, y = p / WIN;
    atomicAdd(&s_cnt[bk], 1);
    atomicMin(&s_xmin[bk], x); atomicMax(&s_xmax[bk], x);
    atomicMin(&s_ymin[bk], y); atomicMax(&s_ymax[bk], y);
  }
  __syncthreads();
  if (t < KCLS && s_cnt[t] > 0) {
    atomicAdd(&ws[t], s_cnt[t]);
    atomicMin(&ws[20 + t], s_xmin[t]); atomicMax(&ws[40 + t], s_xmax[t]);
    atomicMin(&ws[60 + t], s_ymin[t]); atomicMax(&ws[80 + t], s_ymax[t]);
  }
}

// ---------------- box ranking + selection (tiny, 1 thread) ----------------
__global__ void k_select(int* __restrict__ ws) {
  if (threadIdx.x != 0 || blockIdx.x != 0) return;
  int ids[KCLS]; int m = 0;
  for (int i = 0; i < KCLS; ++i) if (ws[i] > 0) ids[m++] = i;
  int nbx = m > 0 ? m - 1 : 0;          // np.unique -> drop first present id
  int cnts[KCLS], rr[KCLS];
  for (int j = 0; j < nbx; ++j) cnts[j] = ws[ids[j + 1]];
  for (int j = 0; j < nbx; ++j) {       // rank = argsort(argsort(-counts)) (stable)
    int r = 0;
    for (int k = 0; k < nbx; ++k)
      if (cnts[k] > cnts[j] || (cnts[k] == cnts[j] && k < j)) r++;
    rr[j] = r;
  }
  int sel[NB - 2]; int ns = 0;
  for (int idx = 0; idx < nbx; ++idx)
    for (int j = 0; j < nbx; ++j)
      if (rr[j] == idx && cnts[j] > COUNT_THRESH && ns < NB - 2) { sel[ns++] = j; break; }
  const int bb[5][4] = {
      {WF / 4, HF / 4, 3 * WF / 4, 3 * HF / 4},
      {0, 0, WF / 3, HF},
      {0, 0, WF, HF / 3},
      {2 * WF / 3, 0, WF, HF},
      {0, 2 * HF / 3, WF, HF}};
  for (int g = 0; g < NB; ++g) {
    int x0, y0, x1, y1;
    if (g < ns) {
      int id = ids[sel[g] + 1];
      x0 = ws[20 + id] >> 4; x1 = ws[40 + id] >> 4;   // boxes / 16 then int-trunc
      y0 = ws[60 + id] >> 4; y1 = ws[80 + id] >> 4;
    } else {
      int i = g - ns;
      x0 = bb[i][0]; y0 = bb[i][1]; x1 = bb[i][2]; y1 = bb[i][3];
    }
    ws[100 + g * 4 + 0] = x0;
    ws[100 + g * 4 + 1] = y0;
    ws[100 + g * 4 + 2] = x1 - x0;   // iw >= 1 (count>10000 guarantees span > 16px)
    ws[100 + g * 4 + 3] = y1 - y0;   // ih >= 1
  }
}

// ---------------- pool_x passthrough + scalar outputs ----------------
__global__ void k_pool(const float* __restrict__ px, float* __restrict__ out) {
  int i = blockIdx.x * blockDim.x + threadIdx.x;
  if (i < NBATCH * CCH) out[i] = px[i];
  if (i == 0) { out[NBATCH * CCH] = (float)NB; out[NBATCH * CCH + 1] = (float)NBATCH; }
}

// ---------------- bilinear crop-resize for slabs 0..19 (slab = n*5+g) ----------------
__global__ void k_resize(const float* __restrict__ feat, const int* __restrict__ ws,
                         float* __restrict__ out) {
  size_t e = (size_t)blockIdx.x * blockDim.x + threadIdx.x;
  if (e >= RESIZE_ELEMS) return;
  int s = (int)(e / SLAB);
  int r = (int)(e % SLAB);
  int n = s / NB, g = s % NB;
  int x0 = ws[100 + g * 4 + 0], y0 = ws[100 + g * 4 + 1];
  int iw = ws[100 + g * 4 + 2], ih = ws[100 + g * 4 + 3];
  int c  = r / (HF * WF);
  int yy = (r % (HF * WF)) / WF;
  int xx = r % WF;
  // jax.image.resize 'bilinear' (upscale => antialias inert): half-pixel centers + edge clamp
  float fy = ((float)yy + 0.5f) * ((float)ih / (float)HF) - 0.5f;
  float fx = ((float)xx + 0.5f) * ((float)iw / (float)WF) - 0.5f;
  fy = fminf(fmaxf(fy, 0.0f), (float)(ih - 1));
  fx = fminf(fmaxf(fx, 0.0f), (float)(iw - 1));
  int iy0 = (int)fy, ix0 = (int)fx;
  float wy = fy - (float)iy0, wx = fx - (float)ix0;
  int iy1 = min(iy0 + 1, ih - 1), ix1 = min(ix0 + 1, iw - 1);
  const float* fb = feat + (size_t)n * SLAB + (size_t)c * (HF * WF);
  float v00 = fb[(y0 + iy0) * WF + x0 + ix0];
  float v01 = fb[(y0 + iy0) * WF + x0 + ix1];
  float v10 = fb[(y0 + iy1) * WF + x0 + ix0];
  float v11 = fb[(y0 + iy1) * WF + x0 + ix1];
  float v = (1.0f - wy) * ((1.0f - wx) * v00 + wx * v01) +
            wy * ((1.0f - wx) * v10 + wx * v11);
  out[OUT_X_OFF + e] = v;
}

// ---------------- feat verbatim copy (slabs 20..23) via Tensor Data Mover ----------------
// One wave per 32KB chunk: TDM global->LDS, s_wait_tensorcnt, TDM LDS->global.
__global__ void __launch_bounds__(32) k_copyfeat(const float* __restrict__ src,
                                                 float* __restrict__ dst) {
  __shared__ float tile[CHUNK];
#if defined(__AMDGCN__) && __has_builtin(__builtin_amdgcn_tensor_load_to_lds) && \
    __has_builtin(__builtin_amdgcn_tensor_store_from_lds) &&                     \
    __has_builtin(__builtin_amdgcn_s_wait_tensorcnt)
  uint32_t lds_off = (uint32_t)(uintptr_t)(&tile[0]);   // flat addr low 32b = LDS offset
  size_t elt = (size_t)blockIdx.x * CHUNK;
  uint64_t sa = (uint64_t)(uintptr_t)(src + elt);
  uint64_t da = (uint64_t)(uintptr_t)(dst + elt);
  // D# group0: count=1 | lds_addr | global_addr[31:0] | global_addr[56:32] + type=2
  v4u g0l = {1u, lds_off, (uint32_t)sa,
             ((uint32_t)(sa >> 32) & 0x01FFFFFFu) | (2u << 30)};
  v4u g0s = {1u, lds_off, (uint32_t)da,
             ((uint32_t)(da >> 32) & 0x01FFFFFFu) | (2u << 30)};
  // D# group1: wg_mask=0, data_size=2 (4B); tensor_dim0=tile_dim0=CHUNK (1-D tile)
  v8i g1 = {(int)(2u << 16),                 // d0: data_size = 4 bytes
            (int)((unsigned)CHUNK << 16),    // d1[31:16] = tensor_dim0[15:0]
            (int)((unsigned)CHUNK >> 16),    // d2[15:0]  = tensor_dim0[31:16]
            (int)((unsigned)CHUNK << 16),    // d3[31:16] = tile_dim0
            0,                               // d4: tile_dim1/2 = 0 (unused -> 1-D)
            (int)CHUNK,                      // d5: tensor_dim0_stride[31:0]
            0, 0};
  v4i z4 = {0, 0, 0, 0};
  v8i z8 = {0, 0, 0, 0, 0, 0, 0, 0};
  // clang-23 / therock-10.0 lane: 6-arg form (v4u, v8i, v4i, v4i, v8i, i32 cpol)
  __builtin_amdgcn_tensor_load_to_lds(g0l, g1, z4, z4, z8, 0);
  __builtin_amdgcn_s_wait_tensorcnt(0);     // LDS tile complete before store descriptor
  __builtin_amdgcn_tensor_store_from_lds(g0s, g1, z4, z4, z8, 0);
  __builtin_amdgcn_s_wait_tensorcnt(0);
#else
  size_t base = (size_t)blockIdx.x * CHUNK;
  const float2* s2 = (const float2*)(src + base);
  float2* d2 = (float2*)(dst + base);       // dst offset is 8-byte aligned
  for (int i = threadIdx.x; i < CHUNK / 2; i += 32) d2[i] = s2[i];
  (void)tile;
#endif
}

extern "C" void kernel_launch(void* const* d_in, const int* in_sizes, int n_in,
                              void* d_out, int out_size, void* d_ws, size_t ws_size,
                              hipStream_t stream) {
  const float* seg  = (const float*)d_in[0];  // (4,20,480,640)
  const float* feat = (const float*)d_in[1];  // (4,512,30,40)
  const float* px   = (const float*)d_in[2];  // (4,512)
  float* out = (float*)d_out;
  int* ws = (int*)d_ws;

  k_init<<<1, 64, 0, stream>>>(ws);
  k_stats<<<NPIX / 256, 256, 0, stream>>>(seg, ws);
  k_select<<<1, 1, 0, stream>>>(ws);
  k_pool<<<(NBATCH * CCH + 255) / 256, 256, 0, stream>>>(px, out);
  k_resize<<<(unsigned)(RESIZE_ELEMS / 256), 256, 0, stream>>>(feat, ws, out);
  k_copyfeat<<<(unsigned)(COPY_FLOATS / CHUNK), 32, 0, stream>>>(
      feat, out + OUT_X_OFF + RESIZE_ELEMS);
}